// MLPKANlayer_19129784336542
// MI455X (gfx1250) — compile-verified
//
#include <hip/hip_runtime.h>
#include <hip/hip_bf16.h>

#define BATCH   2048
#define IN_SZ   128
#define OUT_SZ  128
#define HID     5
#define BBLK    256            // batch elements per subnet block
#define LDS_STRIDE (IN_SZ + 1) // pad to avoid LDS bank conflicts

typedef float v2f __attribute__((ext_vector_type(2)));
typedef float v8f __attribute__((ext_vector_type(8)));

// ---------------------------------------------------------------------------
// Kernel 1: residual GEMM  out[b,o] = sum_i x[b,i] * rs[i*128+o]
// X: [2048,128] fp32, RS: [128,128] fp32 (row-major == flat residual_scaling)
// One wave (32 threads) computes one 16x16 tile via v_wmma_f32_16x16x4_f32.
// ---------------------------------------------------------------------------
__global__ __launch_bounds__(32) void residual_wmma_kernel(
    const float* __restrict__ X,
    const float* __restrict__ RS,
    float* __restrict__ Out)
{
    const int lane  = threadIdx.x;      // 0..31
    const int lo    = lane & 15;
    const int hi    = lane >> 4;        // 0: K pair {0,1}; 1: K pair {2,3}
    const int baseN = blockIdx.x * 16;  // output-column tile (8 tiles)
    const int baseM = blockIdx.y * 16;  // batch-row tile   (128 tiles)

    v8f acc = {};

    #pragma unroll 4
    for (int k = 0; k < IN_SZ; k += 4) {
        // A fragment: 16x4 tile of X. ISA layout (32-bit A 16x4):
        //   lanes 0-15 : VGPR0=K0, VGPR1=K1 ; lanes 16-31: VGPR0=K2, VGPR1=K3
        const float* ap = X + (baseM + lo) * IN_SZ + k + 2 * hi;
        v2f a;
        a.x = ap[0];
        a.y = ap[1];

        // B fragment: 4x16 tile of RS (K rows striped across lanes, mirrored)
        const float* bp = RS + (k + 2 * hi) * OUT_SZ + baseN + lo;
        v2f b;
        b.x = bp[0];
        b.y = bp[OUT_SZ];

        // 8 args: (neg_a, A, neg_b, B, c_mod, C, reuse_a, reuse_b)
        acc = __builtin_amdgcn_wmma_f32_16x16x4_f32(
            false, a, false, b, (short)0, acc, false, false);
    }

    // C/D layout: lane 0-15 -> N=lane, VGPR r -> M=r; lanes 16-31 -> M=8+r
    float* op = Out + (baseM + 8 * hi) * OUT_SZ + baseN + lo;
    #pragma unroll
    for (int r = 0; r < 8; ++r)
        op[r * OUT_SZ] = acc[r];
}

// ---------------------------------------------------------------------------
// Kernel 2: per-net tiny MLPs, accumulated over i on top of the residual GEMM.
//   out[b,o] += sum_i ss[n] * MLP_n(x[b,i]),  n = i*128 + o
// Lanes span batch -> all weight addresses are wave-uniform (scalar SMEM
// loads); x staged in LDS with padded row stride so the inner loop reads conflict-free.
// ---------------------------------------------------------------------------
__device__ __forceinline__ float silu_f(float z) {
    // z * sigmoid(z) = z / (1 + exp(-z)); exp via native v_exp_f32 (2^x)
    float e = __builtin_amdgcn_exp2f(-1.44269504f * z);
    return z * __builtin_amdgcn_rcpf(1.0f + e);
}

__global__ __launch_bounds__(BBLK) void subnet_kernel(
    const float* __restrict__ x,   // [2048,128]
    const float* __restrict__ w0,  // [N,5,1]
    const float* __restrict__ b0,  // [N,5,1]
    const float* __restrict__ w1,  // [N,5,5]
    const float* __restrict__ b1,  // [N,5,1]
    const float* __restrict__ w2,  // [N,1,5]
    const float* __restrict__ b2,  // [N,1,1]
    const float* __restrict__ ss,  // [N]
    float* __restrict__ out)       // [2048,128] (+=)
{
    extern __shared__ float xs[];  // [BBLK][LDS_STRIDE]

    const int o  = blockIdx.x;          // output index, block-uniform
    const int bb = blockIdx.y * BBLK;   // batch base
    const int t  = threadIdx.x;

    // Stage x[bb .. bb+255][0..127] into LDS: coalesced global reads,
    // contiguous LDS writes; padded stride makes the i-loop reads
    // (stride 129 across lanes) bank-conflict free.
    for (int j = t; j < BBLK * IN_SZ; j += BBLK) {
        int bl = j >> 7;          // j / 128
        int i  = j & (IN_SZ - 1); // j % 128
        xs[bl * LDS_STRIDE + i] = x[(bb + bl) * IN_SZ + i];
    }
    __syncthreads();

    const int b = bb + t;
    float acc = 0.0f;

    for (int i = 0; i < IN_SZ; ++i) {
        const int n = i * OUT_SZ + o;   // net index, wave-uniform
        const float xv = xs[t * LDS_STRIDE + i];

        // All weight pointers are uniform -> scalar loads
        const float* W0 = w0 + n * HID;
        const float* B0 = b0 + n * HID;
        const float* W1 = w1 + n * HID * HID;
        const float* B1 = b1 + n * HID;
        const float* W2 = w2 + n * HID;

        float h0[HID], h1[HID];
        #pragma unroll
        for (int h = 0; h < HID; ++h)
            h0[h] = silu_f(fmaf(W0[h], xv, B0[h]));

        #pragma unroll
        for (int h = 0; h < HID; ++h) {
            float s = B1[h];
            #pragma unroll
            for (int k = 0; k < HID; ++k)
                s = fmaf(W1[h * HID + k], h0[k], s);
            h1[h] = silu_f(s);
        }

        float y = b2[n];
        #pragma unroll
        for (int k = 0; k < HID; ++k)
            y = fmaf(W2[k], h1[k], y);

        acc = fmaf(y, ss[n], acc);
    }

    // Residual term was written by the WMMA kernel; accumulate on top.
    out[b * OUT_SZ + o] += acc;
}

// ---------------------------------------------------------------------------
extern "C" void kernel_launch(void* const* d_in, const int* in_sizes, int n_in,
                              void* d_out, int out_size, void* d_ws, size_t ws_size,
                              hipStream_t stream) {
    const float* x   = (const float*)d_in[0];
    const float* w0  = (const float*)d_in[1];
    const float* b0  = (const float*)d_in[2];
    const float* w1  = (const float*)d_in[3];
    const float* b1  = (const float*)d_in[4];
    const float* w2  = (const float*)d_in[5];
    const float* b2  = (const float*)d_in[6];
    const float* ss  = (const float*)d_in[7];
    const float* rs  = (const float*)d_in[8];
    float* out = (float*)d_out;

    // 1) Residual GEMM via WMMA: fully overwrites d_out.
    dim3 gridG(OUT_SZ / 16, BATCH / 16);   // (8, 128) tiles
    residual_wmma_kernel<<<gridG, 32, 0, stream>>>(x, rs, out);

    // 2) Per-net MLPs accumulate on top.
    dim3 gridS(OUT_SZ, BATCH / BBLK);      // (128, 8)
    size_t ldsBytes = (size_t)BBLK * LDS_STRIDE * sizeof(float); // 132 KB
    subnet_kernel<<<gridS, BBLK, ldsBytes, stream>>>(
        x, w0, b0, w1, b1, w2, b2, ss, out);
}